// GraphSageNet_80633716015161
// MI455X (gfx1250) — compile-verified
//
#include <hip/hip_runtime.h>
#include <hip/hip_bf16.h>

// ---------------------------------------------------------------------------
// GraphSage forward on gfx1250 (MI455X), fp32 throughout.
//  - GEMMs via V_WMMA_F32_16X16X4_F32 (exact fp32 reference numerics).
//  - Edge aggregation via once-per-call CSR build + L2-resident gather
//    (no fp atomics in the hot path; graph topology is constant across
//    the 4 layers so the CSR is amortized).
// ---------------------------------------------------------------------------

typedef float v2f __attribute__((ext_vector_type(2)));
typedef float v8f __attribute__((ext_vector_type(8)));

#define HID 128
#define NB_STATS 512  // blocks for norm_relu_stats (partial-sum slots)

// ---------------------------------------------------------------------------
// utility
// ---------------------------------------------------------------------------
__global__ void zero_i32(int* __restrict__ p, int n) {
  for (int i = blockIdx.x * blockDim.x + threadIdx.x; i < n;
       i += gridDim.x * blockDim.x)
    p[i] = 0;
}

// histogram of in-degrees (int atomics, deterministic result)
__global__ void hist_deg(const int* __restrict__ dst, int* __restrict__ cnt,
                         int E) {
  for (int e = blockIdx.x * blockDim.x + threadIdx.x; e < E;
       e += gridDim.x * blockDim.x)
    atomicAdd(&cnt[dst[e]], 1);
}

__global__ void invdeg_from_cnt(const int* __restrict__ cnt,
                                float* __restrict__ invdeg, int N) {
  for (int i = blockIdx.x * blockDim.x + threadIdx.x; i < N;
       i += gridDim.x * blockDim.x)
    invdeg[i] = 1.0f / fmaxf((float)cnt[i], 1.0f);
}

// single-block exclusive prefix scan (Hillis-Steele over 1024-chunks)
__global__ void __launch_bounds__(1024) scan_excl(const int* __restrict__ cnt,
                                                  int* __restrict__ rowptr,
                                                  int N) {
  __shared__ int tmp[1024];
  __shared__ int carry;
  const int t = threadIdx.x;
  if (t == 0) carry = 0;
  __syncthreads();
  for (int base = 0; base < N; base += 1024) {
    int i = base + t;
    int v = (i < N) ? cnt[i] : 0;
    tmp[t] = v;
    __syncthreads();
#pragma unroll
    for (int off = 1; off < 1024; off <<= 1) {
      int addv = (t >= off) ? tmp[t - off] : 0;
      __syncthreads();
      tmp[t] += addv;
      __syncthreads();
    }
    int incl = tmp[t];
    int ctotal = tmp[1023];
    int cbase = carry;
    if (i < N) rowptr[i] = cbase + incl - v;  // exclusive
    __syncthreads();
    if (t == 0) carry = cbase + ctotal;
    __syncthreads();
  }
  if (t == 0) rowptr[N] = carry;
}

// fill CSR slots: eidx[slot] = src of each incoming edge of dst
__global__ void fill_csr(const int* __restrict__ src,
                         const int* __restrict__ dst,
                         const int* __restrict__ rowptr,
                         int* __restrict__ cur, int* __restrict__ eidx,
                         int E) {
  for (int e = blockIdx.x * blockDim.x + threadIdx.x; e < E;
       e += gridDim.x * blockDim.x) {
    int d = dst[e];
    int slot = rowptr[d] + atomicAdd(&cur[d], 1);
    eidx[slot] = src[e];
  }
}

// ---------------------------------------------------------------------------
// CSR gather: c[node] = inv_deg[node] * sum_{s in nbrs(node)} h[s]
// 128 threads = one feature each; neighbor indices staged in LDS; h rows
// are L2-resident (20 MB) so the random gather runs at L2 bandwidth.
// ---------------------------------------------------------------------------
__global__ void __launch_bounds__(128) aggregate_csr(
    const float* __restrict__ h, const int* __restrict__ rowptr,
    const int* __restrict__ eidx, const float* __restrict__ invdeg,
    float* __restrict__ c, int N) {
  __shared__ int idxs[128];
  const int t = threadIdx.x;
  for (int node = blockIdx.x; node < N; node += gridDim.x) {
    int beg = rowptr[node];
    int cnt = rowptr[node + 1] - beg;
    float acc = 0.0f;
    for (int chunk = 0; chunk < cnt; chunk += 128) {
      int m = min(128, cnt - chunk);
      if (t < m) idxs[t] = eidx[beg + chunk + t];
      __syncthreads();
      int i = 0;
      for (; i + 4 <= m; i += 4) {  // clause 4 independent gathers
        float v0 = h[idxs[i + 0] * HID + t];
        float v1 = h[idxs[i + 1] * HID + t];
        float v2 = h[idxs[i + 2] * HID + t];
        float v3 = h[idxs[i + 3] * HID + t];
        acc += v0 + v1 + v2 + v3;
      }
      for (; i < m; ++i) acc += h[idxs[i] * HID + t];
      __syncthreads();
    }
    c[node * HID + t] = acc * invdeg[node];
  }
}

// ---------------------------------------------------------------------------
// Embedding GEMM:  Y[N,128] = X[N,128] @ W[128,128]^T + b
// Block = 256 threads (8 waves); wave w owns 16x16 tile at cols [16w,16w+16).
// ---------------------------------------------------------------------------
__global__ void __launch_bounds__(256) embed_gemm(
    const float* __restrict__ X, const float* __restrict__ W,
    const float* __restrict__ B, float* __restrict__ Y) {
  __shared__ __attribute__((aligned(16))) float tA[16 * HID];  // 8 KB

  const int tid = threadIdx.x;
  const int rowBase = blockIdx.x * 16;

  const float4* Xv = (const float4*)(X + rowBase * HID);
  float4* tAv = (float4*)tA;
  tAv[tid] = Xv[tid];
  tAv[tid + 256] = Xv[tid + 256];
  __syncthreads();

  const int wave = tid >> 5;
  const int lane = tid & 31;
  const int half = lane >> 4;  // 0: K+0/K+1 ; 1: K+2/K+3
  const int l16 = lane & 15;
  const int col = wave * 16 + l16;

  v8f acc;
  {
    float bv = B[col];
#pragma unroll
    for (int r = 0; r < 8; ++r) acc[r] = bv;
  }

  const float* arow = tA + l16 * HID;
  const float* wrow = W + col * HID;

#pragma unroll 4
  for (int k = 0; k < HID; k += 4) {
    const int kk = k + 2 * half;
    v2f a = *(const v2f*)(arow + kk);
    v2f b = *(const v2f*)(wrow + kk);
    acc = __builtin_amdgcn_wmma_f32_16x16x4_f32(false, a, false, b, (short)0,
                                                acc, false, false);
  }

#pragma unroll
  for (int r = 0; r < 8; ++r) {
    int row = rowBase + r + 8 * half;  // lanes 16-31 hold M=r+8
    Y[row * HID + col] = acc[r];
  }
}

// ---------------------------------------------------------------------------
// Node-apply GEMM:  Z[N,128] = [h | c][N,256] @ W[128,256]^T + b
// ---------------------------------------------------------------------------
__global__ void __launch_bounds__(256) layer_gemm(
    const float* __restrict__ h, const float* __restrict__ c,
    const float* __restrict__ W, const float* __restrict__ B,
    float* __restrict__ Z) {
  __shared__ __attribute__((aligned(16))) float tA[16 * 256];  // 16 KB

  const int tid = threadIdx.x;
  const int rowBase = blockIdx.x * 16;

#pragma unroll
  for (int i = 0; i < 2; ++i) {
    int v = tid + i * 256;  // float4 index 0..511
    int r = v >> 5;
    int c4 = v & 31;
    ((float4*)(tA + r * 256))[c4] =
        ((const float4*)(h + (rowBase + r) * HID))[c4];
  }
#pragma unroll
  for (int i = 0; i < 2; ++i) {
    int v = tid + i * 256;
    int r = v >> 5;
    int c4 = v & 31;
    ((float4*)(tA + r * 256 + HID))[c4] =
        ((const float4*)(c + (rowBase + r) * HID))[c4];
  }
  __syncthreads();

  const int wave = tid >> 5;
  const int lane = tid & 31;
  const int half = lane >> 4;
  const int l16 = lane & 15;
  const int col = wave * 16 + l16;

  v8f acc;
  {
    float bv = B[col];
#pragma unroll
    for (int r = 0; r < 8; ++r) acc[r] = bv;
  }

  const float* arow = tA + l16 * 256;
  const float* wrow = W + col * 256;

#pragma unroll 4
  for (int k = 0; k < 256; k += 4) {
    const int kk = k + 2 * half;
    v2f a = *(const v2f*)(arow + kk);
    v2f b = *(const v2f*)(wrow + kk);
    acc = __builtin_amdgcn_wmma_f32_16x16x4_f32(false, a, false, b, (short)0,
                                                acc, false, false);
  }

#pragma unroll
  for (int r = 0; r < 8; ++r) {
    int row = rowBase + r + 8 * half;
    Z[row * HID + col] = acc[r];
  }
}

// ---------------------------------------------------------------------------
// Row L2-normalize + ReLU (in place); per-block BN partials (no atomics).
// ---------------------------------------------------------------------------
__global__ void __launch_bounds__(128) norm_relu_stats(
    float* __restrict__ z, float* __restrict__ part, int N) {
  __shared__ float red[4];
  const int t = threadIdx.x;
  const int w = t >> 5;
  const int lane = t & 31;

  float ps = 0.0f, pss = 0.0f;
  for (int node = blockIdx.x; node < N; node += gridDim.x) {
    float v = z[node * HID + t];
    float sq = v * v;
#pragma unroll
    for (int off = 16; off > 0; off >>= 1) sq += __shfl_xor(sq, off, 32);
    if (lane == 0) red[w] = sq;
    __syncthreads();
    float tot = red[0] + red[1] + red[2] + red[3];
    __syncthreads();
    float nrm = fmaxf(sqrtf(tot), 1e-12f);
    float r = fmaxf(v / nrm, 0.0f);
    z[node * HID + t] = r;
    ps += r;
    pss += r * r;
  }
  part[blockIdx.x * 256 + t] = ps;
  part[blockIdx.x * 256 + 128 + t] = pss;
}

// ---------------------------------------------------------------------------
// BN finalize: reduce partials, bn[j] = gamma/sqrt(var+eps),
//              bn[128+j] = beta - mu*bn[j]
// ---------------------------------------------------------------------------
__global__ void __launch_bounds__(256) bn_finalize(
    const float* __restrict__ part, int nb, const float* __restrict__ gamma,
    const float* __restrict__ beta, float* __restrict__ bn, float invN) {
  __shared__ float tot[256];
  const int j = threadIdx.x;
  float s = 0.0f;
  for (int b = 0; b < nb; ++b) s += part[b * 256 + j];
  tot[j] = s;
  __syncthreads();
  if (j < 128) {
    float mu = tot[j] * invN;
    float var = tot[128 + j] * invN - mu * mu;
    float a = gamma[j] / sqrtf(var + 1e-5f);
    bn[j] = a;
    bn[128 + j] = beta[j] - mu * a;
  }
}

// ---------------------------------------------------------------------------
// out = h_in + (z*bn_a + bn_b)
// ---------------------------------------------------------------------------
__global__ void residual_bn(const float* __restrict__ h,
                            const float* __restrict__ z,
                            const float* __restrict__ bn,
                            float* __restrict__ out, int total) {
  for (int i = blockIdx.x * blockDim.x + threadIdx.x; i < total;
       i += gridDim.x * blockDim.x) {
    int j = i & (HID - 1);
    out[i] = h[i] + z[i] * bn[j] + bn[HID + j];
  }
}

// ---------------------------------------------------------------------------
// launch
// ---------------------------------------------------------------------------
extern "C" void kernel_launch(void* const* d_in, const int* in_sizes, int n_in,
                              void* d_out, int out_size, void* d_ws,
                              size_t ws_size, hipStream_t stream) {
  const float* h0 = (const float*)d_in[0];     // [N,128]
  const int* src = (const int*)d_in[1];        // [E]
  const int* dst = (const int*)d_in[2];        // [E]
  const float* W_emb = (const float*)d_in[3];  // [128,128]
  const float* b_emb = (const float*)d_in[4];  // [128]
  const float* Ws = (const float*)d_in[5];     // [4,128,256]
  const float* bs = (const float*)d_in[6];     // [4,128]
  const float* gam = (const float*)d_in[7];    // [4,128]
  const float* bet = (const float*)d_in[8];    // [4,128]

  const int N = in_sizes[0] / HID;  // 40000
  const int E = in_sizes[1];        // 640000
  const int NF = N * HID;
  const int L = 4;

  // ---- workspace layout ----
  float* ws = (float*)d_ws;
  float* H = ws;                       // N*128
  float* C = H + NF;                   // N*128
  float* Z = C + NF;                   // N*128
  float* DEG = Z + NF;                 // N      (inv_deg)
  float* BN = DEG + N;                 // 256
  float* PART = BN + 256;              // NB_STATS*256
  int* rowptr = (int*)(PART + NB_STATS * 256);  // N+1
  int* cnt = rowptr + (N + 1);         // N  (histogram, then cursor)
  int* eidx = cnt + N;                 // E

  const int mtiles = N / 16;  // 2500

  // ---- CSR build (topology constant across layers) ----
  zero_i32<<<256, 256, 0, stream>>>(cnt, N);
  hist_deg<<<1024, 256, 0, stream>>>(dst, cnt, E);
  invdeg_from_cnt<<<256, 256, 0, stream>>>(cnt, DEG, N);
  scan_excl<<<1, 1024, 0, stream>>>(cnt, rowptr, N);
  zero_i32<<<256, 256, 0, stream>>>(cnt, N);  // reuse as cursor
  fill_csr<<<1024, 256, 0, stream>>>(src, dst, rowptr, cnt, eidx, E);

  // ---- input embedding ----
  embed_gemm<<<mtiles, 256, 0, stream>>>(h0, W_emb, b_emb, H);

  // ---- layers ----
  for (int l = 0; l < L; ++l) {
    aggregate_csr<<<2048, 128, 0, stream>>>(H, rowptr, eidx, DEG, C, N);
    layer_gemm<<<mtiles, 256, 0, stream>>>(H, C, Ws + l * HID * 256,
                                           bs + l * HID, Z);
    norm_relu_stats<<<NB_STATS, 128, 0, stream>>>(Z, PART, N);
    bn_finalize<<<1, 256, 0, stream>>>(PART, NB_STATS, gam + l * HID,
                                       bet + l * HID, BN, 1.0f / (float)N);
    float* out = (l == L - 1) ? (float*)d_out : H;
    residual_bn<<<4096, 256, 0, stream>>>(H, Z, BN, out, NF);
  }
}